// Attention_21449066676384
// MI455X (gfx1250) — compile-verified
//
#include <hip/hip_runtime.h>
#include <math.h>

typedef float v2f __attribute__((ext_vector_type(2)));
typedef float v8f __attribute__((ext_vector_type(8)));

// D = A(16x4,f32) * B(4x16,f32) + C(16x16,f32), full fp32 matrix op on CDNA5
#define WMMA_F32(a, b, c) \
  __builtin_amdgcn_wmma_f32_16x16x4_f32(false, (a), false, (b), (short)0, (c), false, false)

// ---------------------------------------------------------------------------
// GEMM: C = A(M,K) @ B(K,N) [+ bias], all row-major fp32.
// 256 threads = 8 wave32 waves. Block tile 128(M) x 128(N); wave tile 64x32.
// K stepped by 16 (4 WMMA k-steps). Software-pipelined: next panel is
// prefetched into registers while the WMMA chains run on the current panel.
// Pitches: A=18 (distinct banks for frag reads, 8B-aligned float2),
//          B=136 (2*136 % 64 == 16 -> lane halves hit disjoint bank groups).
// ---------------------------------------------------------------------------
__global__ __launch_bounds__(256) void gemm_f32_wmma(
    const float* __restrict__ A, const float* __restrict__ B,
    const float* __restrict__ bias, float* __restrict__ C,
    int M, int K, int N)
{
  __shared__ float As[128 * 18];
  __shared__ float Bs[16 * 136];

  const int tid  = threadIdx.x;
  const int lane = tid & 31;
  const int w    = tid >> 5;
  const int ln   = lane & 15;        // index within 16-lane group (M/N dim)
  const int tg2  = (lane >> 4) << 1; // K sub-offset from lane half
  const int wm   = w & 1;            // wave M position (0/1 -> +0/+64)
  const int wn   = w >> 1;           // wave N position (0..3 -> +wn*32)
  const int bm   = blockIdx.y * 128;
  const int bn   = blockIdx.x * 128;

  // Per-thread staging coordinates (constant over k-steps)
  int arow[4], acp[4], brow[4], bcp[4];
  #pragma unroll
  for (int i = 0; i < 4; i++) {
    int f = tid + i * 256;
    arow[i] = f >> 3;  acp[i] = (f & 7) << 1;
    brow[i] = f >> 6;  bcp[i] = (f & 63) << 1;
  }

  v8f acc[4][2] = {};

  // Prologue: prefetch panel k0=0 into registers
  v2f ga[4], gb[4];
  #pragma unroll
  for (int i = 0; i < 4; i++) {
    ga[i] = *(const v2f*)&A[(size_t)(bm + arow[i]) * K + acp[i]];
    gb[i] = *(const v2f*)&B[(size_t)brow[i] * N + bn + bcp[i]];
  }

  for (int k0 = 0; k0 < K; k0 += 16) {
    // Commit prefetched panel to LDS
    #pragma unroll
    for (int i = 0; i < 4; i++) {
      *(v2f*)&As[arow[i] * 18  + acp[i]] = ga[i];
      *(v2f*)&Bs[brow[i] * 136 + bcp[i]] = gb[i];
    }
    __syncthreads();

    // Issue next panel's global loads; they complete under the WMMA chains
    if (k0 + 16 < K) {
      #pragma unroll
      for (int i = 0; i < 4; i++) {
        ga[i] = *(const v2f*)&A[(size_t)(bm + arow[i]) * K + (k0 + 16) + acp[i]];
        gb[i] = *(const v2f*)&B[(size_t)(k0 + 16 + brow[i]) * N + bn + bcp[i]];
      }
    }

    #pragma unroll
    for (int kk = 0; kk < 4; kk++) {
      v2f af[4];
      v2f bf[2];
      #pragma unroll
      for (int mi = 0; mi < 4; mi++)
        af[mi] = *(v2f*)&As[(wm * 64 + mi * 16 + ln) * 18 + kk * 4 + tg2];
      #pragma unroll
      for (int ni = 0; ni < 2; ni++) {
        bf[ni][0] = Bs[(kk * 4 + tg2    ) * 136 + wn * 32 + ni * 16 + ln];
        bf[ni][1] = Bs[(kk * 4 + tg2 + 1) * 136 + wn * 32 + ni * 16 + ln];
      }
      // 8 independent accumulator chains -> hides WMMA RAW hazard latency
      #pragma unroll
      for (int mi = 0; mi < 4; mi++)
        #pragma unroll
        for (int ni = 0; ni < 2; ni++)
          acc[mi][ni] = WMMA_F32(af[mi], bf[ni], acc[mi][ni]);
    }
    __syncthreads();
  }

  // Store: C/D layout -> lane ln is column, VGPR r is row r + 8*(lane>=16)
  const int rofs = (lane >> 4) << 3;
  #pragma unroll
  for (int mi = 0; mi < 4; mi++) {
    #pragma unroll
    for (int ni = 0; ni < 2; ni++) {
      int col = bn + wn * 32 + ni * 16 + ln;
      float bv = bias ? bias[col] : 0.0f;
      #pragma unroll
      for (int r = 0; r < 8; r++) {
        int row = bm + wm * 64 + mi * 16 + r + rofs;
        C[(size_t)row * N + col] = acc[mi][ni][r] + bv;
      }
    }
  }
}

// ---------------------------------------------------------------------------
// Flash-style attention, fp32 WMMA throughout.
// Grid (16,16,4) = (query-slab, head, batch). 128 threads = 4 waves,
// each wave owns 16 query rows. Software-pipelined: V(kb) loads run under
// the S=QK^T WMMA chain; K(kb+1)+mask loads run under the P.V WMMA chain.
// Two barriers per key block:
//   barrier A (top):   kbuf/mb staged; also fences prev iter's vbuf/pbuf reads
//   barrier B (mid):   vbuf + pbuf staged for the PV phase
// ---------------------------------------------------------------------------
#define KLD 68   // 4*KLD % 64 spreads frag-read rows over all banks; 16B rows
#define VLD 72   // 2*VLD % 64 == 16 -> lane halves hit disjoint bank groups
#define PLD 68

__global__ __launch_bounds__(128) void attn_f32_wmma(
    const float* __restrict__ qkv,   // (B*N, 3072): [q | k | v], head-major
    const float* __restrict__ mask,  // (B, N)
    float* __restrict__ out)         // (B*N, 1024)
{
  __shared__ float kbuf[64 * KLD];
  __shared__ float vbuf[64 * VLD];
  __shared__ float pbuf[4][16 * PLD];
  __shared__ float mb[64];

  const int N = 1024, C3 = 3072;
  const int tid  = threadIdx.x;
  const int lane = tid & 31;
  const int w    = tid >> 5;
  const int ln   = lane & 15;
  const int hl   = lane >> 4;
  const int tg2  = hl << 1;
  const int qt   = blockIdx.x;   // query slab 0..15
  const int h    = blockIdx.y;   // head
  const int b    = blockIdx.z;   // batch
  const int qbase = qt * 64 + w * 16;
  const float scale = 0.125f;    // D=64 -> 1/sqrt(64)

  // Preload Q fragments (A-matrix layout): lane holds row qbase+ln,
  // d = 4*kk + tg2 + {0,1}  -> one float2 per k-step, kept in registers.
  const float* qrow = qkv + (size_t)(b * N + qbase + ln) * C3 + h * 64;
  v2f qf[16];
  #pragma unroll
  for (int kk = 0; kk < 16; kk++)
    qf[kk] = *(const v2f*)(qrow + kk * 4 + tg2);

  // Per-thread K/V staging coordinates (8 float4 each, coalesced)
  int sr[8], sc[8];
  #pragma unroll
  for (int i = 0; i < 8; i++) {
    int f = tid + i * 128;
    sr[i] = f >> 4;
    sc[i] = (f & 15) << 2;
  }
  const float* kvp = qkv + (size_t)(b * N) * C3 + 1024 + h * 64;  // K section

  v8f o[4] = {};
  float mi[8], li[8];
  #pragma unroll
  for (int r = 0; r < 8; r++) { mi[r] = -1e30f; li[r] = 0.0f; }

  // Prologue: prefetch K(0) and mask(0)
  float4 kreg[8], vreg[8];
  #pragma unroll
  for (int i = 0; i < 8; i++)
    kreg[i] = *(const float4*)(kvp + (size_t)sr[i] * C3 + sc[i]);
  float mval = 0.0f;
  if (tid < 64) mval = (1.0f - mask[b * N + tid]) * -5.0f;

  for (int kb = 0; kb < 16; kb++) {
    const int kbase = kb * 64;

    // Commit K tile + mask bias to LDS
    #pragma unroll
    for (int i = 0; i < 8; i++)
      *(float4*)&kbuf[sr[i] * KLD + sc[i]] = kreg[i];
    if (tid < 64) mb[tid] = mval;
    __syncthreads();  // barrier A

    // Issue V(kb) loads; they complete under the S-phase WMMA chain
    #pragma unroll
    for (int i = 0; i < 8; i++)
      vreg[i] = *(const float4*)(kvp + 1024 + (size_t)(kbase + sr[i]) * C3 + sc[i]);

    // S = Q @ K^T : B-frag of K^T is a K row (key=ln), contiguous float2 in d
    v8f s[4] = {};
    #pragma unroll
    for (int kk = 0; kk < 16; kk++) {
      #pragma unroll
      for (int kt = 0; kt < 4; kt++) {
        v2f kf = *(v2f*)&kbuf[(kt * 16 + ln) * KLD + kk * 4 + tg2];
        s[kt] = WMMA_F32(qf[kk], kf, s[kt]);
      }
    }

    // Online softmax. C/D layout: s[kt][r] belongs to row r+8*hl, col ln
    // within each 16-key tile -> row reductions are 16-lane butterflies.
    float mnew[8], alpha[8];
    #pragma unroll
    for (int r = 0; r < 8; r++) {
      float mx = -1e30f;
      #pragma unroll
      for (int kt = 0; kt < 4; kt++) {
        float v = s[kt][r] * scale + mb[kt * 16 + ln];
        s[kt][r] = v;
        mx = fmaxf(mx, v);
      }
      #pragma unroll
      for (int off = 1; off < 16; off <<= 1)
        mx = fmaxf(mx, __shfl_xor(mx, off, 32));
      mnew[r]  = fmaxf(mi[r], mx);
      alpha[r] = __expf(mi[r] - mnew[r]);
      mi[r]    = mnew[r];
    }
    #pragma unroll
    for (int r = 0; r < 8; r++) {
      float sum = 0.0f;
      #pragma unroll
      for (int kt = 0; kt < 4; kt++) {
        float p = __expf(s[kt][r] - mnew[r]);
        s[kt][r] = p;
        sum += p;
      }
      #pragma unroll
      for (int off = 1; off < 16; off <<= 1)
        sum += __shfl_xor(sum, off, 32);
      li[r] = li[r] * alpha[r] + sum;
      #pragma unroll
      for (int dt = 0; dt < 4; dt++) o[dt][r] *= alpha[r];
    }

    // Stage V tile and P strip (C/D -> A layout conversion through LDS)
    #pragma unroll
    for (int i = 0; i < 8; i++)
      *(float4*)&vbuf[sr[i] * VLD + sc[i]] = vreg[i];
    #pragma unroll
    for (int kt = 0; kt < 4; kt++)
      #pragma unroll
      for (int r = 0; r < 8; r++)
        pbuf[w][(r + hl * 8) * PLD + kt * 16 + ln] = s[kt][r];
    __syncthreads();  // barrier B

    // Issue K(kb+1)+mask loads; they complete under the PV WMMA chain
    if (kb + 1 < 16) {
      #pragma unroll
      for (int i = 0; i < 8; i++)
        kreg[i] = *(const float4*)(kvp + (size_t)(kbase + 64 + sr[i]) * C3 + sc[i]);
      if (tid < 64) mval = (1.0f - mask[b * N + kbase + 64 + tid]) * -5.0f;
    }

    // O += P @ V
    #pragma unroll
    for (int kk = 0; kk < 16; kk++) {
      v2f pf = *(v2f*)&pbuf[w][ln * PLD + kk * 4 + tg2];
      #pragma unroll
      for (int dt = 0; dt < 4; dt++) {
        v2f vf;
        vf[0] = vbuf[(kk * 4 + tg2    ) * VLD + dt * 16 + ln];
        vf[1] = vbuf[(kk * 4 + tg2 + 1) * VLD + dt * 16 + ln];
        o[dt] = WMMA_F32(pf, vf, o[dt]);
      }
    }
  }

  // Normalize by softmax denominator and store (B,N,H,D) -> (B*N, 1024)
  #pragma unroll
  for (int r = 0; r < 8; r++) {
    float inv = 1.0f / li[r];
    int row = b * N + qbase + r + hl * 8;
    #pragma unroll
    for (int dt = 0; dt < 4; dt++)
      out[(size_t)row * 1024 + h * 64 + dt * 16 + ln] = o[dt][r] * inv;
  }
}

// ---------------------------------------------------------------------------
extern "C" void kernel_launch(void* const* d_in, const int* in_sizes, int n_in,
                              void* d_out, int out_size, void* d_ws, size_t ws_size,
                              hipStream_t stream) {
  const float* x      = (const float*)d_in[0];
  const float* mask   = (const float*)d_in[1];
  const float* W_qkv  = (const float*)d_in[2];
  const float* W_proj = (const float*)d_in[3];
  const float* b_proj = (const float*)d_in[4];
  float* out = (float*)d_out;

  float* qkv     = (float*)d_ws;                    // 4096 x 3072 f32 (48 MB)
  float* attnout = qkv + (size_t)4096 * 3072;       // 4096 x 1024 f32 (16 MB)

  // 1) qkv = x @ W_qkv                (4096,1024) x (1024,3072)
  gemm_f32_wmma<<<dim3(3072 / 128, 4096 / 128), 256, 0, stream>>>(
      x, W_qkv, nullptr, qkv, 4096, 1024, 3072);

  // 2) attention per (batch, head, 64-query slab)
  attn_f32_wmma<<<dim3(16, 16, 4), 128, 0, stream>>>(qkv, mask, attnout);

  // 3) out = attnout @ W_proj + b_proj   (4096,1024) x (1024,1024)
  gemm_f32_wmma<<<dim3(1024 / 128, 4096 / 128), 256, 0, stream>>>(
      attnout, W_proj, b_proj, out, 4096, 1024, 1024);
}